// GaussianModel_53231824666916
// MI455X (gfx1250) — compile-verified
//
#include <hip/hip_runtime.h>
#include <stdint.h>

// ---------------------------------------------------------------------------
// cov = (R(q_normalized) * diag(s)) * (R * diag(s))^T  per point, N = 4M.
// Memory-bound: 112 MB in + 144 MB out @ 23.3 TB/s ~= 11 us floor.
// CDNA5 data-movement strategy:
//   scale  (12 B/pt)  : async global->LDS DMA of contiguous 3072 B slab
//   rot    (16 B/pt)  : coalesced b128 non-temporal loads straight to VGPRs
//   cov    (36 B/pt)  : stage in LDS (stride-9 dwords, conflict-free), then
//                       async LDS->global DMA (no VGPR round trip)
// WMMA deliberately unused: 3x3 per-point products batched into 16x16 tiles
// waste >80% of MAC lanes + shuffle cost, and the kernel is bandwidth-bound.
// ---------------------------------------------------------------------------

typedef float v4f __attribute__((ext_vector_type(4)));

#define CHUNK 256  // points per block (== blockDim.x); 4M % 256 == 0

__device__ __forceinline__ void qs_to_cov(v4f q, float sx, float sy, float sz,
                                          float c[9]) {
  // reference ordering: q = [w, x, y, z]
  float w = q.x, x = q.y, y = q.z, z = q.w;
  float n2  = w * w + x * x + y * y + z * z;
  float inv = 1.0f / sqrtf(n2);  // exact path to match reference numerics
  w *= inv; x *= inv; y *= inv; z *= inv;

  float xx = x * x, yy = y * y, zz = z * z;
  float xy = x * y, xz = x * z, yz = y * z;
  float wx = w * x, wy = w * y, wz = w * z;

  float r00 = 1.f - 2.f * (yy + zz), r01 = 2.f * (xy - wz), r02 = 2.f * (xz + wy);
  float r10 = 2.f * (xy + wz), r11 = 1.f - 2.f * (xx + zz), r12 = 2.f * (yz - wx);
  float r20 = 2.f * (xz - wy), r21 = 2.f * (yz + wx), r22 = 1.f - 2.f * (xx + yy);

  // M = R * diag(s)  =>  cov[i][k] = sum_j R[i][j] R[k][j] s[j]^2 (symmetric)
  float s0 = sx * sx, s1 = sy * sy, s2 = sz * sz;
  float m00 = r00 * r00 * s0 + r01 * r01 * s1 + r02 * r02 * s2;
  float m01 = r00 * r10 * s0 + r01 * r11 * s1 + r02 * r12 * s2;
  float m02 = r00 * r20 * s0 + r01 * r21 * s1 + r02 * r22 * s2;
  float m11 = r10 * r10 * s0 + r11 * r11 * s1 + r12 * r12 * s2;
  float m12 = r10 * r20 * s0 + r11 * r21 * s1 + r12 * r22 * s2;
  float m22 = r20 * r20 * s0 + r21 * r21 * s1 + r22 * r22 * s2;

  c[0] = m00; c[1] = m01; c[2] = m02;
  c[3] = m01; c[4] = m11; c[5] = m12;
  c[6] = m02; c[7] = m12; c[8] = m22;
}

__global__ __launch_bounds__(CHUNK) void gaussian_cov_kernel(
    const float* __restrict__ rot,    // (N,4)
    const float* __restrict__ scale,  // (N,3)
    float* __restrict__ out,          // (N,3,3)
    long long N) {
  __shared__ __align__(16) float s_scale[CHUNK * 3];  // 3072 B
  __shared__ __align__(16) float s_out[CHUNK * 9];    // 9216 B

  const int t = threadIdx.x;
  const long long base = (long long)blockIdx.x * CHUNK;
  const bool full = (base + CHUNK) <= N;

  if (full) {
    // ---- 1) Async DMA the contiguous 3072 B scale slab into LDS ----------
    // 192 lanes x global_load_async_to_lds_b128 (tracked by ASYNCcnt).
    if (t < (CHUNK * 3) / 4) {
      uint64_t gsrc = (uint64_t)(uintptr_t)(scale + base * 3 + (long long)t * 4);
      unsigned lds_off = (unsigned)(uintptr_t)&s_scale[t * 4];  // low 32 = LDS offset
      asm volatile("global_load_async_to_lds_b128 %0, %1, off"
                   :: "v"(lds_off), "v"(gsrc) : "memory");
    }

    // ---- 2) Coalesced b128 NT load of the quaternion (streamed once) -----
    v4f q = __builtin_nontemporal_load((const v4f*)rot + (base + t));

    asm volatile("s_wait_asynccnt 0" ::: "memory");
    __syncthreads();

    // ---- 3) Compute; stride-3 / stride-9 LDS (both coprime w/ 64 banks) --
    float sx = s_scale[t * 3 + 0];
    float sy = s_scale[t * 3 + 1];
    float sz = s_scale[t * 3 + 2];

    float c[9];
    qs_to_cov(q, sx, sy, sz, c);

#pragma unroll
    for (int k = 0; k < 9; ++k) s_out[t * 9 + k] = c[k];

    __syncthreads();

    // ---- 4) Async LDS->global DMA of the 9216 B output slab --------------
    // 576 x b128: no VGPR round trip; S_ENDPGM's implicit wait-idle drains
    // the outstanding ASYNCcnt, and LDS persists until the workgroup ends.
    float* og = out + base * 9;  // base*9*4 B is 16B-aligned
#pragma unroll
    for (int idx = t; idx < (CHUNK * 9) / 4; idx += CHUNK) {
      uint64_t gdst = (uint64_t)(uintptr_t)(og + (long long)idx * 4);
      unsigned lds_off = (unsigned)(uintptr_t)&s_out[idx * 4];
      asm volatile("global_store_async_from_lds_b128 %0, %1, off"
                   :: "v"(gdst), "v"(lds_off) : "memory");
    }
  } else {
    // ---- Tail chunk (not hit for N = 4M): guarded scalar path ------------
    long long n = base + t;
    if (n < N) {
      v4f q = ((const v4f*)rot)[n];
      float c[9];
      qs_to_cov(q, scale[n * 3 + 0], scale[n * 3 + 1], scale[n * 3 + 2], c);
#pragma unroll
      for (int k = 0; k < 9; ++k) out[n * 9 + k] = c[k];
    }
  }
}

extern "C" void kernel_launch(void* const* d_in, const int* in_sizes, int n_in,
                              void* d_out, int out_size, void* d_ws, size_t ws_size,
                              hipStream_t stream) {
  (void)n_in; (void)out_size; (void)d_ws; (void)ws_size;
  const float* rotation = (const float*)d_in[0];  // (N,4) f32
  const float* scale    = (const float*)d_in[1];  // (N,3) f32
  float* out            = (float*)d_out;          // (N,3,3) f32

  long long N = (long long)in_sizes[0] / 4;
  long long nblocks = (N + CHUNK - 1) / CHUNK;

  gaussian_cov_kernel<<<(dim3)(unsigned)nblocks, CHUNK, 0, stream>>>(
      rotation, scale, out, N);
}